// TextGNN_2456721293450
// MI455X (gfx1250) — compile-verified
//
#include <hip/hip_runtime.h>
#include <hip/hip_bf16.h>
#include <math.h>

// GAT layer for MI455X (gfx1250, wave32).
// h = x@W via V_WMMA_F32_16X16X4_F32 with 4Mx2N register blocking;
// edge softmax via L2-resident atomics (h = 51MB fits the 192MB L2).

#define F_IN 128
#define NHEAD 4
#define CH 32
#define HC 128
#define NEG_SLOPE 0.2f

typedef __attribute__((ext_vector_type(2))) float v2f;
typedef __attribute__((ext_vector_type(8))) float v8f;

// ---------------- init: zero out/denom, amax = -inf ----------------
__global__ void gat_init(float* __restrict__ out, float* __restrict__ amax,
                         float* __restrict__ denom, int n_nodes) {
    int idx = blockIdx.x * blockDim.x + threadIdx.x;
    if (idx < n_nodes * HC) out[idx] = 0.0f;
    if (idx < n_nodes * NHEAD) {
        amax[idx] = -INFINITY;
        denom[idx] = 0.0f;
    }
}

// ---------------- GEMM: H = X[N,128] @ W[128,128] via f32 WMMA ----------------
// One wave computes a 64x32 output block: 4 M-tiles x 2 N-tiles (8 x v8f acc).
// Per K-step (K+=4): 4 unconditional b64 A-loads + 4 b32 B-loads feed 8 WMMAs.
// Layouts per CDNA5 ISA 7.12.2:
//   A lanes 0-15: M=lane, v0=K0,v1=K1 ; lanes 16-31: M=lane-16, v0=K2,v1=K3
//   B lanes 0-15: N=lane, v0=K0,v1=K1 ; lanes 16-31: N=lane-16, v0=K2,v1=K3
//   D vgpr v: lanes 0-15 -> (M=v,   N=lane), lanes 16-31 -> (M=v+8, N=lane-16)
__global__ void gat_gemm_wmma(const float* __restrict__ X, const float* __restrict__ W,
                              float* __restrict__ Hm, int n_nodes) {
    const int tm = blockIdx.x >> 2;           // 64-row M block
    const int tn = blockIdx.x & 3;            // 32-col N block
    const int m0 = tm * 64, n0 = tn * 32;
    const int lane = threadIdx.x & 31;
    const int half = lane >> 4;               // selects K pair {0,1} vs {2,3}
    const int r = lane & 15;

    // Clamped row indices -> loads are always in-bounds and unconditional.
    const int rmax = n_nodes - 1;
    int row0 = m0 + 0  + r; row0 = row0 < rmax ? row0 : rmax;
    int row1 = m0 + 16 + r; row1 = row1 < rmax ? row1 : rmax;
    int row2 = m0 + 32 + r; row2 = row2 < rmax ? row2 : rmax;
    int row3 = m0 + 48 + r; row3 = row3 < rmax ? row3 : rmax;

    const float* __restrict__ xp0 = X + (size_t)row0 * F_IN + 2 * half;
    const float* __restrict__ xp1 = X + (size_t)row1 * F_IN + 2 * half;
    const float* __restrict__ xp2 = X + (size_t)row2 * F_IN + 2 * half;
    const float* __restrict__ xp3 = X + (size_t)row3 * F_IN + 2 * half;
    const float* __restrict__ bp  = W + (size_t)(2 * half) * HC + n0 + r;

    v8f acc00 = {}, acc01 = {}, acc10 = {}, acc11 = {};
    v8f acc20 = {}, acc21 = {}, acc30 = {}, acc31 = {};

#pragma unroll 4
    for (int k = 0; k < F_IN; k += 4) {
        v2f a0 = *(const v2f*)(xp0 + k);
        v2f a1 = *(const v2f*)(xp1 + k);
        v2f a2 = *(const v2f*)(xp2 + k);
        v2f a3 = *(const v2f*)(xp3 + k);
        const float* bk = bp + (size_t)k * HC;
        v2f b0, b1;
        b0.x = bk[0];       b0.y = bk[HC];
        b1.x = bk[16];      b1.y = bk[HC + 16];
        acc00 = __builtin_amdgcn_wmma_f32_16x16x4_f32(false, a0, false, b0, (short)0, acc00, false, false);
        acc01 = __builtin_amdgcn_wmma_f32_16x16x4_f32(false, a0, false, b1, (short)0, acc01, false, false);
        acc10 = __builtin_amdgcn_wmma_f32_16x16x4_f32(false, a1, false, b0, (short)0, acc10, false, false);
        acc11 = __builtin_amdgcn_wmma_f32_16x16x4_f32(false, a1, false, b1, (short)0, acc11, false, false);
        acc20 = __builtin_amdgcn_wmma_f32_16x16x4_f32(false, a2, false, b0, (short)0, acc20, false, false);
        acc21 = __builtin_amdgcn_wmma_f32_16x16x4_f32(false, a2, false, b1, (short)0, acc21, false, false);
        acc30 = __builtin_amdgcn_wmma_f32_16x16x4_f32(false, a3, false, b0, (short)0, acc30, false, false);
        acc31 = __builtin_amdgcn_wmma_f32_16x16x4_f32(false, a3, false, b1, (short)0, acc31, false, false);
    }

    // Store 64x32 block; guard rows only (tail M block).
    v8f accs[4][2] = {{acc00, acc01}, {acc10, acc11}, {acc20, acc21}, {acc30, acc31}};
#pragma unroll
    for (int mt = 0; mt < 4; ++mt) {
#pragma unroll
        for (int v = 0; v < 8; ++v) {
            int row = m0 + mt * 16 + v + 8 * half;
            if (row < n_nodes) {
                float* op = Hm + (size_t)row * HC + n0 + r;
                op[0]  = accs[mt][0][v];
                op[16] = accs[mt][1][v];
            }
        }
    }
}

// ---------------- per-(node,head) attention score dots ----------------
__global__ void gat_scores(const float* __restrict__ Hm, const float* __restrict__ att,
                           float* __restrict__ Si, float* __restrict__ Sj, int n_nodes) {
    int idx = blockIdx.x * blockDim.x + threadIdx.x;  // n*4 + h
    if (idx >= n_nodes * NHEAD) return;
    int n = idx >> 2, hh = idx & 3;
    const float* __restrict__ hp = Hm + (size_t)n * HC + hh * CH;
    const float* __restrict__ ai = att + hh * (2 * CH);
    const float* __restrict__ aj = ai + CH;
    float si = 0.0f, sj = 0.0f;
#pragma unroll
    for (int c = 0; c < CH; ++c) {
        float hv = hp[c];
        si = fmaf(hv, ai[c], si);
        sj = fmaf(hv, aj[c], sj);
    }
    Si[idx] = si;
    Sj[idx] = sj;
}

// ---------------- float atomic max via int/uint bit trick ----------------
__device__ __forceinline__ void atomic_max_f32(float* addr, float v) {
    if (v >= 0.0f)
        atomicMax((int*)addr, __float_as_int(v));
    else
        atomicMin((unsigned int*)addr, __float_as_uint(v));
}

__device__ __forceinline__ float leaky(float x) {
    return x > 0.0f ? x : NEG_SLOPE * x;
}

// ---------------- pass A: segment max of logits ----------------
// Virtual edges: e < E -> (i=src[e], j=dst[e], skip if i==j); e >= E -> self loop.
__global__ void gat_edge_max(const int* __restrict__ src, const int* __restrict__ dst,
                             const float* __restrict__ Si, const float* __restrict__ Sj,
                             float* __restrict__ amax, int n_edges, int n_nodes) {
    int e = blockIdx.x * blockDim.x + threadIdx.x;
    if (e >= n_edges + n_nodes) return;
    int i, j;
    if (e < n_edges) {
        i = src[e]; j = dst[e];
        if (i == j) return;
    } else {
        i = j = e - n_edges;
    }
    const float4 si = *(const float4*)(Si + (size_t)i * NHEAD);
    const float4 sj = *(const float4*)(Sj + (size_t)j * NHEAD);
    float* ap = amax + (size_t)i * NHEAD;
    atomic_max_f32(ap + 0, leaky(si.x + sj.x));
    atomic_max_f32(ap + 1, leaky(si.y + sj.y));
    atomic_max_f32(ap + 2, leaky(si.z + sj.z));
    atomic_max_f32(ap + 3, leaky(si.w + sj.w));
}

// ---------------- pass B: exp weights, denom sum, unnormalized scatter ----------------
// One wave32 per virtual edge; lane -> (head = lane>>3, 4 channels = (lane&7)*4).
__global__ void gat_edge_scatter(const int* __restrict__ src, const int* __restrict__ dst,
                                 const float* __restrict__ Hm,
                                 const float* __restrict__ Si, const float* __restrict__ Sj,
                                 const float* __restrict__ amax,
                                 float* __restrict__ denom, float* __restrict__ out,
                                 int n_edges, int n_nodes) {
    int wid = blockIdx.x * (blockDim.x >> 5) + (threadIdx.x >> 5);
    int lane = threadIdx.x & 31;
    if (wid >= n_edges + n_nodes) return;
    int i, j;
    if (wid < n_edges) {
        i = src[wid]; j = dst[wid];
        if (i == j) return;
    } else {
        i = j = wid - n_edges;
    }
    const int hh = lane >> 3;
    const int c0 = (lane & 7) << 2;
    float a = leaky(Si[(size_t)i * NHEAD + hh] + Sj[(size_t)j * NHEAD + hh]);
    float ex = __expf(a - amax[(size_t)i * NHEAD + hh]);
    if ((lane & 7) == 0) atomicAdd(denom + (size_t)i * NHEAD + hh, ex);
    const float4 hv = *(const float4*)(Hm + (size_t)j * HC + hh * CH + c0);
    float* op = out + (size_t)i * HC + hh * CH + c0;
    atomicAdd(op + 0, ex * hv.x);
    atomicAdd(op + 1, ex * hv.y);
    atomicAdd(op + 2, ex * hv.z);
    atomicAdd(op + 3, ex * hv.w);
}

// ---------------- pass C: normalize + bias ----------------
__global__ void gat_finalize(float* __restrict__ out, const float* __restrict__ denom,
                             const float* __restrict__ bias, int n_nodes) {
    int idx = blockIdx.x * blockDim.x + threadIdx.x;
    if (idx >= n_nodes * HC) return;
    int n = idx >> 7, f = idx & (HC - 1), hh = f >> 5;
    float d = denom[(size_t)n * NHEAD + hh];
    d = d > 1e-16f ? d : 1e-16f;
    out[idx] = out[idx] / d + bias[f];
}

extern "C" void kernel_launch(void* const* d_in, const int* in_sizes, int n_in,
                              void* d_out, int out_size, void* d_ws, size_t ws_size,
                              hipStream_t stream) {
    const float* x    = (const float*)d_in[0];
    const int*   ei   = (const int*)d_in[1];
    const float* w    = (const float*)d_in[2];
    const float* att  = (const float*)d_in[3];
    const float* bias = (const float*)d_in[4];
    float* out = (float*)d_out;

    const int n_nodes = in_sizes[0] / F_IN;   // 100000
    const int n_edges = in_sizes[1] / 2;      // 1600000
    const int* src = ei;                      // edge_index[0] = segment (row) index
    const int* dst = ei + n_edges;            // edge_index[1] = neighbor (col) index

    // workspace layout (floats): H[N*128] | Si[N*4] | Sj[N*4] | amax[N*4] | denom[N*4]
    float* Hm    = (float*)d_ws;
    float* Si    = Hm   + (size_t)n_nodes * HC;
    float* Sj    = Si   + (size_t)n_nodes * NHEAD;
    float* amax  = Sj   + (size_t)n_nodes * NHEAD;
    float* denom = amax + (size_t)n_nodes * NHEAD;

    const int tot_out = n_nodes * HC;
    gat_init<<<(tot_out + 255) / 256, 256, 0, stream>>>(out, amax, denom, n_nodes);

    const int mblocks = (n_nodes + 63) / 64;
    gat_gemm_wmma<<<mblocks * 4, 32, 0, stream>>>(x, w, Hm, n_nodes);

    gat_scores<<<(n_nodes * NHEAD + 255) / 256, 256, 0, stream>>>(Hm, att, Si, Sj, n_nodes);

    const int ev = n_edges + n_nodes;
    gat_edge_max<<<(ev + 255) / 256, 256, 0, stream>>>(src, dst, Si, Sj, amax,
                                                       n_edges, n_nodes);
    gat_edge_scatter<<<(ev + 7) / 8, 256, 0, stream>>>(src, dst, Hm, Si, Sj, amax,
                                                       denom, out, n_edges, n_nodes);
    gat_finalize<<<(tot_out + 255) / 256, 256, 0, stream>>>(out, denom, bias, n_nodes);
}